// GAT_15161234555290
// MI455X (gfx1250) — compile-verified
//
#include <hip/hip_runtime.h>

#define GAT_N     100000
#define GAT_E     1600000
#define GAT_HEADS 4
#define NEG_SLOPE 0.2f

typedef __attribute__((ext_vector_type(2))) float v2f;
typedef __attribute__((ext_vector_type(8))) float v8f;

// ---------------------------------------------------------------------------
// GEMM: Y[N,Fout] = X[N,Fin] @ W[Fin,Fout], fp32 via V_WMMA_F32_16X16X4_F32.
// Block = 128 threads = 4 waves; each wave owns one 16x16 output tile.
// Fragment layout (ISA 7.12.2, wave32):
//   A 16x4:  lane l holds row M=l&15, regs j=0,1 hold K = j + 2*(l>>4)
//   B 4x16:  lane l holds col N=l&15, regs j=0,1 hold K = j + 2*(l>>4)
//   C 16x16: lane l holds col N=l&15, regs r=0..7 hold M = r + 8*(l>>4)
// ---------------------------------------------------------------------------
__global__ __launch_bounds__(128)
void gat_gemm_f32_wmma(const float* __restrict__ X, const float* __restrict__ W,
                       float* __restrict__ Y, int Fin, int Fout) {
  const int lane = threadIdx.x & 31;
  const int wave = threadIdx.x >> 5;
  const int row0 = blockIdx.x * 16;
  const int col0 = (blockIdx.y * 4 + wave) * 16;
  const int lm   = lane & 15;
  const int half = lane >> 4;

  v8f acc = {};
  const float* __restrict__ xrow = X + (size_t)(row0 + lm) * Fin + 2 * half;
  const float* __restrict__ wcol = W + col0 + lm;

  for (int k0 = 0; k0 < Fin; k0 += 4) {
    v2f a, b;
    a.x = xrow[k0];
    a.y = xrow[k0 + 1];
    const int kb = k0 + 2 * half;
    b.x = wcol[(size_t)kb * Fout];
    b.y = wcol[(size_t)(kb + 1) * Fout];
    acc = __builtin_amdgcn_wmma_f32_16x16x4_f32(false, a, false, b,
                                                (short)0, acc, false, false);
  }

#pragma unroll
  for (int r = 0; r < 8; ++r) {
    Y[(size_t)(row0 + r + 8 * half) * Fout + col0 + lm] = acc[r];
  }
}

// ---------------------------------------------------------------------------
// el[n,h] = sum_d H[n,h,d]*al[h,d] ; er likewise. One thread per (node, head).
// ---------------------------------------------------------------------------
__global__ void gat_attn_coef(const float* __restrict__ H, const float* __restrict__ al,
                              const float* __restrict__ ar, float* __restrict__ el,
                              float* __restrict__ er, int Hd) {
  int i = blockIdx.x * blockDim.x + threadIdx.x;
  if (i >= GAT_N * GAT_HEADS) return;
  const int node = i / GAT_HEADS;
  const int h    = i - node * GAT_HEADS;
  const float* __restrict__ hp  = H + (size_t)node * (GAT_HEADS * Hd) + h * Hd;
  const float* __restrict__ alp = al + h * Hd;
  const float* __restrict__ arp = ar + h * Hd;
  float sl = 0.f, sr = 0.f;
  for (int d = 0; d < Hd; ++d) {
    float v = hp[d];
    sl += v * alp[d];
    sr += v * arp[d];
  }
  el[i] = sl;
  er[i] = sr;
}

// Monotonic uint encoding of float for atomic max over signed floats.
__device__ __forceinline__ unsigned f2ord(float f) {
  unsigned u = __float_as_uint(f);
  return (u & 0x80000000u) ? ~u : (u | 0x80000000u);
}
__device__ __forceinline__ float ord2f(unsigned m) {
  return (m & 0x80000000u) ? __uint_as_float(m ^ 0x80000000u) : __uint_as_float(~m);
}

__device__ __forceinline__ float leaky(float v) {
  return v > 0.f ? v : NEG_SLOPE * v;
}

// Pass 1: per-destination segment max of leaky-relu'd logits.
__global__ void gat_edge_max(const int* __restrict__ src, const int* __restrict__ dst,
                             const float* __restrict__ el, const float* __restrict__ er,
                             unsigned* __restrict__ mord) {
  int e = blockIdx.x * blockDim.x + threadIdx.x;
  if (e >= GAT_E) return;
  const int s = src[e], d = dst[e];
#pragma unroll
  for (int h = 0; h < GAT_HEADS; ++h) {
    float v = leaky(el[s * GAT_HEADS + h] + er[d * GAT_HEADS + h]);
    atomicMax(&mord[d * GAT_HEADS + h], f2ord(v));
  }
}

// Pass 2: ee = exp(e - m[dst]); denom[dst] += ee.
__global__ void gat_edge_exp(const int* __restrict__ src, const int* __restrict__ dst,
                             const float* __restrict__ el, const float* __restrict__ er,
                             const unsigned* __restrict__ mord, float* __restrict__ ee,
                             float* __restrict__ denom) {
  int e = blockIdx.x * blockDim.x + threadIdx.x;
  if (e >= GAT_E) return;
  const int s = src[e], d = dst[e];
#pragma unroll
  for (int h = 0; h < GAT_HEADS; ++h) {
    float v = leaky(el[s * GAT_HEADS + h] + er[d * GAT_HEADS + h]);
    float m = ord2f(mord[d * GAT_HEADS + h]);
    float x = expf(v - m);
    ee[(size_t)e * GAT_HEADS + h] = x;
    atomicAdd(&denom[d * GAT_HEADS + h], x);
  }
}

// Pass 3: agg[dst] += (ee/denom[dst]) * H[src]. One thread per (edge, 4 feats).
__global__ void gat_aggregate(const int* __restrict__ src, const int* __restrict__ dst,
                              const float* __restrict__ H, const float* __restrict__ ee,
                              const float* __restrict__ denom, float* __restrict__ agg,
                              int Fout, int Hd) {
  const int chunks = Fout >> 2;                 // float4 chunks per row
  int tid = blockIdx.x * blockDim.x + threadIdx.x;
  if (tid >= GAT_E * chunks) return;
  const int e = tid / chunks;
  const int c = tid - e * chunks;
  const int h = (c << 2) / Hd;                  // head for this chunk
  const int s = src[e], d = dst[e];
  const float alpha = ee[(size_t)e * GAT_HEADS + h] / denom[d * GAT_HEADS + h];
  const float4 hv = *(const float4*)(H + (size_t)s * Fout + (c << 2));
  float* __restrict__ ap = agg + (size_t)d * Fout + (c << 2);
  atomicAdd(ap + 0, alpha * hv.x);
  atomicAdd(ap + 1, alpha * hv.y);
  atomicAdd(ap + 2, alpha * hv.z);
  atomicAdd(ap + 3, alpha * hv.w);
}

// Hidden-layer epilogue (in place): x = elu(agg + b).
__global__ void gat_epilogue_elu(float* __restrict__ agg, const float* __restrict__ b,
                                 int Fout) {
  int i = blockIdx.x * blockDim.x + threadIdx.x;
  if (i >= GAT_N * Fout) return;
  const int f = i % Fout;
  float v = agg[i] + b[f];
  agg[i] = v > 0.f ? v : expm1f(v);
}

// Output-layer epilogue: out[n,d] = mean_h(agg[n,h,d] + b[h,d]).
__global__ void gat_final_mean(const float* __restrict__ agg, const float* __restrict__ b,
                               float* __restrict__ out, int Hd) {
  int i = blockIdx.x * blockDim.x + threadIdx.x;
  if (i >= GAT_N * Hd) return;
  const int node = i / Hd;
  const int d    = i - node * Hd;
  float sum = 0.f;
#pragma unroll
  for (int h = 0; h < GAT_HEADS; ++h)
    sum += agg[(size_t)node * (GAT_HEADS * Hd) + h * Hd + d] + b[h * Hd + d];
  out[i] = sum * (1.0f / GAT_HEADS);
}

// ---------------------------------------------------------------------------
// One GAT layer (everything except the activation/mean epilogue).
// ---------------------------------------------------------------------------
static void run_gat_layer(const float* x, const float* W, const float* al, const float* ar,
                          const int* src, const int* dst,
                          float* hbuf, float* agg, float* el, float* er,
                          unsigned* mord, float* denom, float* ee,
                          int Fin, int Fout, int Hd, hipStream_t stream) {
  dim3 gg(GAT_N / 16, Fout / 64);
  gat_gemm_f32_wmma<<<gg, 128, 0, stream>>>(x, W, hbuf, Fin, Fout);

  const int nh = GAT_N * GAT_HEADS;
  gat_attn_coef<<<(nh + 255) / 256, 256, 0, stream>>>(hbuf, al, ar, el, er, Hd);

  hipMemsetAsync(mord, 0, (size_t)nh * sizeof(unsigned), stream);
  hipMemsetAsync(denom, 0, (size_t)nh * sizeof(float), stream);
  hipMemsetAsync(agg, 0, (size_t)GAT_N * Fout * sizeof(float), stream);

  gat_edge_max<<<(GAT_E + 255) / 256, 256, 0, stream>>>(src, dst, el, er, mord);
  gat_edge_exp<<<(GAT_E + 255) / 256, 256, 0, stream>>>(src, dst, el, er, mord, ee, denom);

  const int total = GAT_E * (Fout / 4);
  gat_aggregate<<<(total + 255) / 256, 256, 0, stream>>>(src, dst, hbuf, ee, denom, agg,
                                                         Fout, Hd);
}

extern "C" void kernel_launch(void* const* d_in, const int* in_sizes, int n_in,
                              void* d_out, int out_size, void* d_ws, size_t ws_size,
                              hipStream_t stream) {
  const float* nfeat = (const float*)d_in[0];
  const int*   src   = (const int*)d_in[1];
  const int*   dst   = (const int*)d_in[2];
  const float* W0 = (const float*)d_in[3];
  const float* al0 = (const float*)d_in[4];
  const float* ar0 = (const float*)d_in[5];
  const float* b0 = (const float*)d_in[6];
  const float* W1 = (const float*)d_in[7];
  const float* al1 = (const float*)d_in[8];
  const float* ar1 = (const float*)d_in[9];
  const float* b1 = (const float*)d_in[10];
  const float* W2 = (const float*)d_in[11];
  const float* al2 = (const float*)d_in[12];
  const float* ar2 = (const float*)d_in[13];
  const float* b2 = (const float*)d_in[14];
  float* out = (float*)d_out;

  // Workspace layout (all offsets 256B-aligned):
  //   A     : N*256 f32  (GEMM output h)            @ 0
  //   B     : N*256 f32  (aggregation / next input) @ 102,400,000
  //   el,er : N*4 f32 each
  //   mord  : N*4 u32
  //   denom : N*4 f32
  //   ee    : E*4 f32
  char* ws = (char*)d_ws;
  float*    A     = (float*)(ws);
  float*    B     = (float*)(ws + 102400000ull);
  float*    el    = (float*)(ws + 204800000ull);
  float*    er    = (float*)(ws + 206400000ull);
  unsigned* mord  = (unsigned*)(ws + 208000000ull);
  float*    denom = (float*)(ws + 209600000ull);
  float*    ee    = (float*)(ws + 211200000ull);

  // Layer 0: [N,128] -> [N,4,32], elu, flatten -> B
  run_gat_layer(nfeat, W0, al0, ar0, src, dst, A, B, el, er, mord, denom, ee,
                128, 128, 32, stream);
  gat_epilogue_elu<<<(GAT_N * 128 + 255) / 256, 256, 0, stream>>>(B, b0, 128);

  // Layer 1: [N,128] -> [N,4,32], elu, flatten -> B
  run_gat_layer(B, W1, al1, ar1, src, dst, A, B, el, er, mord, denom, ee,
                128, 128, 32, stream);
  gat_epilogue_elu<<<(GAT_N * 128 + 255) / 256, 256, 0, stream>>>(B, b1, 128);

  // Layer 2: [N,128] -> [N,4,64], +bias, mean over heads -> out [N,64]
  run_gat_layer(B, W2, al2, ar2, src, dst, A, B, el, er, mord, denom, ee,
                128, 256, 64, stream);
  gat_final_mean<<<(GAT_N * 64 + 255) / 256, 256, 0, stream>>>(B, b2, out, 64);
}